// HierarchicalGCNPyG_55121610277008
// MI455X (gfx1250) — compile-verified
//
#include <hip/hip_runtime.h>
#include <math.h>

typedef __attribute__((ext_vector_type(2))) float v2f;
typedef __attribute__((ext_vector_type(8))) float v8f;

#define NN 28

#define WMMA4(Af, Bf, Cf) \
  __builtin_amdgcn_wmma_f32_16x16x4_f32(false, (Af), false, (Bf), (short)0, (Cf), false, false)

__constant__ int   c_par[NN] = {-1,0,0,0,0,1,1,2,3,4,4,5,5,6,7,8,9,10,11,12,13,14,14,14,15,15,16,17};
// deg[n] = (#children) + (has parent) + 1 (self loop)
__constant__ float c_deg[NN] = {5.f,4.f,3.f,3.f,4.f,4.f,3.f,3.f,3.f,3.f,3.f,3.f,3.f,3.f,
                                5.f,4.f,3.f,3.f,2.f,2.f,2.f,2.f,2.f,2.f,2.f,2.f,2.f,2.f};

// ---------------------------------------------------------------------------
// Phase 1: Y[B,64] = X[B,256] @ W1[256,64] via V_WMMA_F32_16X16X4_F32.
// One wave computes a 16x64 tile (4 v8f accumulators); 8 waves/block.
// ---------------------------------------------------------------------------
__global__ __launch_bounds__(256) void gemm_xw1(const float* __restrict__ X,
                                                const float* __restrict__ W1,
                                                float* __restrict__ Y, int B) {
  const int lane = threadIdx.x & 31;
  const int wave = threadIdx.x >> 5;
  const int row0 = blockIdx.x * 128 + wave * 16;
  const int kh   = lane >> 4;        // lane half: K pair {2kh, 2kh+1}
  const int col  = lane & 15;
  int m = row0 + (lane & 15);
  if (m > B - 1) m = B - 1;          // clamp loads; WMMA needs EXEC all-ones

  v8f acc0 = {}, acc1 = {}, acc2 = {}, acc3 = {};
  const float* xrow = X + (size_t)m * 256 + 2 * kh;

  for (int k0 = 0; k0 < 256; k0 += 4) {
    __builtin_prefetch(xrow + k0 + 128, 0, 1);     // global_prefetch_b8
    v2f a = *(const v2f*)(xrow + k0);
    const float* w0 = W1 + (size_t)(k0 + 2 * kh) * 64 + col;
    const float* w1 = w0 + 64;
    v2f b0 = { w0[0],  w1[0]  };
    v2f b1 = { w0[16], w1[16] };
    v2f b2 = { w0[32], w1[32] };
    v2f b3 = { w0[48], w1[48] };
    acc0 = WMMA4(a, b0, acc0);
    acc1 = WMMA4(a, b1, acc1);
    acc2 = WMMA4(a, b2, acc2);
    acc3 = WMMA4(a, b3, acc3);
  }
  #pragma unroll
  for (int v = 0; v < 8; ++v) {
    int r = row0 + 8 * kh + v;
    if (r < B) {
      float* yb = Y + (size_t)r * 64 + col;
      yb[0]  = acc0[v];
      yb[16] = acc1[v];
      yb[32] = acc2[v];
      yb[48] = acc3[v];
    }
  }
}

// ---------------------------------------------------------------------------
// Phase 2: batched WMMA tree head. 256 threads = 8 waves, S=8 samples/block.
// LDS layout is node-major [n][s][c] so A-fragments are contiguous float2.
//  - channel GEMMs pack (node-pair x 8 samples) into M=16
//  - node mixings use zero-padded A_norm [32x28] as shared A-matrix, with the
//    flattened (sample,channel) axis as N; bias+ReLU fused into C stores.
// Layer-1 (broadcast collapse) is fused into layer-2's A-fragment generation.
// Requires B % 8 == 0.
// ---------------------------------------------------------------------------
__global__ __launch_bounds__(256) void tree_head(
    const float* __restrict__ Y,
    const float* __restrict__ b1,
    const float* __restrict__ W2, const float* __restrict__ b2,
    const float* __restrict__ W3, const float* __restrict__ b3,
    const float* __restrict__ W4, const float* __restrict__ b4,
    const float* __restrict__ W5, const float* __restrict__ b5,
    float* __restrict__ out, int Btot) {
  __shared__ __align__(16) float sY[8 * 64];        //  2 KB  Y tile
  __shared__ __align__(16) float sZ[NN * 256];      // 28 KB  Z ping  (max 28x8x32)
  __shared__ __align__(16) float sH[NN * 256];      // 28 KB  H pong  (max 28x8x32)
  __shared__ __align__(16) float sApad[32 * NN];    // 3.5 KB A_norm, rows 28..31 = 0
  __shared__ __align__(16) float sW4p[16 * 16];     //  1 KB  W4 padded 16x16
  __shared__ float sS[NN];
  __shared__ float sB1[64];
  __shared__ float sB2[32];
  __shared__ float sB3[16];
  __shared__ float sB4[8];
  __shared__ float sW5[8];
  __shared__ float sB5v;

  const int tid  = threadIdx.x;
  const int lane = tid & 31;
  const int wv   = tid >> 5;
  const int b0   = blockIdx.x * 8;
  const int r15  = lane & 15;      // A-fragment / C-column row id
  const int kh   = lane >> 4;      // lane half: K pair {2kh, 2kh+1}
  const int col  = lane & 15;

  // ---- stage 0: cooperative staging --------------------------------------
  for (int i = tid; i < 8 * 64; i += 256) sY[i] = Y[(size_t)b0 * 64 + i];
  for (int i = tid; i < 32 * NN; i += 256) {
    int n = i / NN, mm = i % NN;
    bool adj = (n < NN) && ((n == mm) || (c_par[mm] == n) || (c_par[n] == mm));
    sApad[i] = adj ? (1.0f / sqrtf(c_deg[n])) * (1.0f / sqrtf(c_deg[mm])) : 0.0f;
  }
  for (int i = tid; i < 16 * 16; i += 256) {
    int r = i >> 4, c = i & 15;
    sW4p[i] = (c < 8) ? W4[r * 8 + c] : 0.0f;
  }
  if (tid < 64) sB1[tid] = b1[tid];
  if (tid < 32) sB2[tid] = b2[tid];
  if (tid < 16) sB3[tid] = b3[tid];
  if (tid < 8)  { sB4[tid] = b4[tid]; sW5[tid] = W5[tid]; }
  if (tid == 0) sB5v = b5[0];
  __syncthreads();
  if (tid < NN) {
    float s = 0.f;
    for (int mm = 0; mm < NN; ++mm) s += sApad[tid * NN + mm];
    sS[tid] = s;
  }
  __syncthreads();

  // ---- stage 1: Z2 = H1 @ W2, H1 fused from Y (28 jobs: 14 pairs x 2 ntile)
  {
    for (int j = wv; j < 28; j += 8) {
      const int p = j >> 1, t = j & 1;
      const int nA = 2 * p + (r15 >> 3);   // node for this A-frag row
      const int sI = r15 & 7;              // sample for this A-frag row
      const float sn = sS[nA];
      v8f c = {};
      #pragma unroll
      for (int k0 = 0; k0 < 64; k0 += 4) {
        const int kk = k0 + 2 * kh;
        v2f y  = *(const v2f*)&sY[sI * 64 + kk];
        v2f bb = *(const v2f*)&sB1[kk];
        v2f a;
        a.x = fmaxf(fmaf(sn, y.x, bb.x), 0.f);
        a.y = fmaxf(fmaf(sn, y.y, bb.y), 0.f);
        v2f bf = { W2[kk * 32 + t * 16 + col], W2[(kk + 1) * 32 + t * 16 + col] };
        c = WMMA4(a, bf, c);
      }
      const int nC = 2 * p + kh;           // C rows: lanes0-15 -> node 2p, 16-31 -> 2p+1
      #pragma unroll
      for (int v = 0; v < 8; ++v) sZ[nC * 256 + v * 32 + t * 16 + col] = c[v];
    }
  }
  __syncthreads();

  // ---- stage 2: H2 = relu(Apad @ Z2 + b2)   D[32 x 256] (32 jobs)
  for (int j = wv; j < 32; j += 8) {
    const int mt = j >> 4, nt = j & 15;
    v8f c = {};
    #pragma unroll
    for (int k0 = 0; k0 < 28; k0 += 4) {
      const int kk = k0 + 2 * kh;
      v2f a  = *(const v2f*)&sApad[(mt * 16 + r15) * NN + kk];
      v2f bf = { sZ[kk * 256 + nt * 16 + col], sZ[(kk + 1) * 256 + nt * 16 + col] };
      c = WMMA4(a, bf, c);
    }
    const int jc = nt * 16 + col;
    const int ch = jc & 31;
    #pragma unroll
    for (int v = 0; v < 8; ++v) {
      int m = mt * 16 + v + 8 * kh;
      if (m < NN) sH[m * 256 + jc] = fmaxf(c[v] + sB2[ch], 0.f);
    }
  }
  __syncthreads();

  // ---- stage 3: Z3 = H2 @ W3   (14 node-pair jobs, K=32, N=16)
  for (int j = wv; j < 14; j += 8) {
    const int nA = 2 * j + (r15 >> 3);
    const int sI = r15 & 7;
    v8f c = {};
    #pragma unroll
    for (int k0 = 0; k0 < 32; k0 += 4) {
      const int kk = k0 + 2 * kh;
      v2f a  = *(const v2f*)&sH[nA * 256 + sI * 32 + kk];
      v2f bf = { W3[kk * 16 + col], W3[(kk + 1) * 16 + col] };
      c = WMMA4(a, bf, c);
    }
    const int nC = 2 * j + kh;
    #pragma unroll
    for (int v = 0; v < 8; ++v) sZ[nC * 128 + v * 16 + col] = c[v];
  }
  __syncthreads();

  // ---- stage 4: H3 = relu(Apad @ Z3 + b3)   D[32 x 128] (16 jobs)
  for (int j = wv; j < 16; j += 8) {
    const int mt = j >> 3, nt = j & 7;
    v8f c = {};
    #pragma unroll
    for (int k0 = 0; k0 < 28; k0 += 4) {
      const int kk = k0 + 2 * kh;
      v2f a  = *(const v2f*)&sApad[(mt * 16 + r15) * NN + kk];
      v2f bf = { sZ[kk * 128 + nt * 16 + col], sZ[(kk + 1) * 128 + nt * 16 + col] };
      c = WMMA4(a, bf, c);
    }
    const int jc = nt * 16 + col;
    const int ch = jc & 15;
    #pragma unroll
    for (int v = 0; v < 8; ++v) {
      int m = mt * 16 + v + 8 * kh;
      if (m < NN) sH[m * 128 + jc] = fmaxf(c[v] + sB3[ch], 0.f);
    }
  }
  __syncthreads();

  // ---- stage 5: Z4 = H3 @ W4pad   (14 node-pair jobs, K=16, N=16 (8 live))
  for (int j = wv; j < 14; j += 8) {
    const int nA = 2 * j + (r15 >> 3);
    const int sI = r15 & 7;
    v8f c = {};
    #pragma unroll
    for (int k0 = 0; k0 < 16; k0 += 4) {
      const int kk = k0 + 2 * kh;
      v2f a  = *(const v2f*)&sH[nA * 128 + sI * 16 + kk];
      v2f bf = { sW4p[kk * 16 + col], sW4p[(kk + 1) * 16 + col] };
      c = WMMA4(a, bf, c);
    }
    const int nC = 2 * j + kh;
    if (col < 8) {
      #pragma unroll
      for (int v = 0; v < 8; ++v) sZ[nC * 64 + v * 8 + col] = c[v];
    }
  }
  __syncthreads();

  // ---- stage 6: H4 = relu(Apad @ Z4 + b4)   D[32 x 64] (8 jobs, 1/wave)
  {
    const int mt = wv >> 2, nt = wv & 3;
    v8f c = {};
    #pragma unroll
    for (int k0 = 0; k0 < 28; k0 += 4) {
      const int kk = k0 + 2 * kh;
      v2f a  = *(const v2f*)&sApad[(mt * 16 + r15) * NN + kk];
      v2f bf = { sZ[kk * 64 + nt * 16 + col], sZ[(kk + 1) * 64 + nt * 16 + col] };
      c = WMMA4(a, bf, c);
    }
    const int jc = nt * 16 + col;
    const int ch = jc & 7;
    #pragma unroll
    for (int v = 0; v < 8; ++v) {
      int m = mt * 16 + v + 8 * kh;
      if (m < NN) sH[m * 64 + jc] = fmaxf(c[v] + sB4[ch], 0.f);
    }
  }
  __syncthreads();

  // ---- stage 7: layer 5 + sibling softmax + path products (per-wave sample)
  float* lz  = &sZ[wv * NN];              // sZ region is free now
  float* lgs = &sZ[256 + wv * NN];
  float* lcp = &sZ[512 + wv * NN];
  const int b = b0 + wv;

  if (lane < NN) {
    float acc = 0.f;
    #pragma unroll
    for (int k = 0; k < 8; ++k) acc = fmaf(sH[lane * 64 + wv * 8 + k], sW5[k], acc);
    lz[lane] = acc;
  }
  __syncthreads();
  if (lane < NN) {
    float acc = sB5v;
    for (int mm = 0; mm < NN; ++mm) acc = fmaf(sApad[lane * NN + mm], lz[mm], acc);
    lgs[lane] = acc;
  }
  __syncthreads();
  if (lane < NN) {
    if (lane == 0) {
      lcp[0] = 0.f;                       // root cond_lp unused by M
    } else {
      int p = c_par[lane];
      float mx = -1e30f;
      for (int mm = 1; mm < NN; ++mm)
        if (c_par[mm] == p) mx = fmaxf(mx, lgs[mm]);
      float ssum = 0.f;
      for (int mm = 1; mm < NN; ++mm)
        if (c_par[mm] == p) ssum += expf(lgs[mm] - mx);
      lcp[lane] = lgs[lane] - mx - logf(ssum);
    }
  }
  __syncthreads();
  if (lane < NN) {
    float acc = 0.f;
    int node = lane;
    while (node != 0) { acc += lcp[node]; node = c_par[node]; }
    out[(size_t)b * NN + lane] = expf(acc);                          // path_probs
    out[(size_t)Btot * NN + (size_t)b * NN + lane] = lgs[lane];      // logits
  }
}

extern "C" void kernel_launch(void* const* d_in, const int* in_sizes, int n_in,
                              void* d_out, int out_size, void* d_ws, size_t ws_size,
                              hipStream_t stream) {
  const float* x  = (const float*)d_in[0];
  const float* W1 = (const float*)d_in[1];
  const float* b1 = (const float*)d_in[2];
  const float* W2 = (const float*)d_in[3];
  const float* b2 = (const float*)d_in[4];
  const float* W3 = (const float*)d_in[5];
  const float* b3 = (const float*)d_in[6];
  const float* W4 = (const float*)d_in[7];
  const float* b4 = (const float*)d_in[8];
  const float* W5 = (const float*)d_in[9];
  const float* b5 = (const float*)d_in[10];
  float* out = (float*)d_out;

  const int B = in_sizes[0] / 256;   // 16384
  float* Y = (float*)d_ws;           // B*64 floats = 4 MB scratch

  gemm_xw1<<<(B + 127) / 128, 256, 0, stream>>>(x, W1, Y, B);
  tree_head<<<B / 8, 256, 0, stream>>>(Y, b1, W2, b2, W3, b3, W4, b4, W5, b5, out, B);
}